// SoftProposal_87771951661250
// MI455X (gfx1250) — compile-verified
//
#include <hip/hip_runtime.h>
#include <math.h>

typedef float v2f __attribute__((ext_vector_type(2)));
typedef float v8f __attribute__((ext_vector_type(8)));

#define NN 14
#define S  196
#define CP 256          // column dim padded to 16 tiles of 16 (branchless)
#define KCH 1024
#define BB 128
#define KC 32
#define MAX_ITER 10
#define ERR_TH 1e-4f

// workspace layout in float units
#define WS_SQ   0
#define WS_D    (WS_SQ + BB*S)            // 25088
#define WS_M0   (WS_D + BB*S*S)           // + 4,917,248
#define WS_M1   (WS_M0 + BB*S)
#define WS_ERR  (WS_M1 + BB*S)            // 128 partials
#define WS_DONE (WS_ERR + BB)             // 11 int flags (reinterpreted)

// ---------------- kernel 1: sq + M0 + done[0] ----------------
__global__ void sp_init(const float* __restrict__ f, float* __restrict__ ws) {
    int b = blockIdx.x, tid = threadIdx.x;
    if (tid < S) {
        const float* fb = f + (size_t)b * KCH * S + tid;
        float acc = 0.f;
        for (int k = 0; k < KCH; ++k) { float v = fb[(size_t)k * S]; acc += v * v; }
        ws[WS_SQ + b * S + tid] = acc;
        ws[WS_M0 + b * S + tid] = 1.0f / (float)S;
    }
    if (b == 0 && tid == 0) ((int*)(ws + WS_DONE))[0] = 0;
}

// ------------- kernel 2: Gram via WMMA f32 + D build -------------
// grid = (13, 128); block = 256 (8 waves, 2 tile-cols each, branchless).
// LDS pair-interleaved layouts: frag loads are single ds_load_b64.
__global__ void sp_gram_d(const float* __restrict__ f, float* __restrict__ ws) {
    extern __shared__ float smem[];
    float* Abuf   = smem;                  // [512 pairs][16 rows][2] = 64 KB
    float* Bbuf   = Abuf + KCH * 16;       // [16 pairs][256 cols][2] = 32 KB
    float* Grow   = Bbuf + KC * CP;        // [16][256]               = 16 KB
    float* rowsum = Grow + 16 * CP;        // [16]

    const int mt  = blockIdx.x;            // tile row 0..12
    const int b   = blockIdx.y;
    const int m0  = mt * 16;
    const int tid = threadIdx.x;
    const int wave = __builtin_amdgcn_readfirstlane(tid >> 5);  // 0..7, scalar
    const int lane = tid & 31;
    const int lh = lane >> 4, l15 = lane & 15;

    const float* fb = f + (size_t)b * KCH * S;
    const float* sq = ws + WS_SQ + b * S;

    // A strip: f[b][k][m0..m0+15], paired layout Abuf[(k>>1)*32 + sl*2 + (k&1)]
    for (int idx = tid; idx < KCH * 16; idx += 256) {
        int k = idx >> 4, sl = idx & 15, s = m0 + sl;
        float v = (s < S) ? fb[(size_t)k * S + s] : 0.f;
        Abuf[((k >> 1) * 16 + sl) * 2 + (k & 1)] = v;
    }

    v8f acc0 = {}, acc1 = {};              // tile cols c0 = 2*wave, c1 = 2*wave+1
    const int c0 = 2 * wave, c1 = 2 * wave + 1;

    for (int k0 = 0; k0 < KCH; k0 += KC) {
        __syncthreads();
        // B chunk: paired layout Bbuf[((kk>>1)*256 + t)*2 + (kk&1)], coalesced global reads
        for (int idx = tid; idx < KC * CP; idx += 256) {
            int kk = idx >> 8, t = idx & 255;
            float v = (t < S) ? fb[(size_t)(k0 + kk) * S + t] : 0.f;
            Bbuf[((kk >> 1) * CP + t) * 2 + (kk & 1)] = v;
        }
        __syncthreads();
        for (int kk = 0; kk < KC; kk += 4) {
            // A fragment: lanes 0-15 carry K=kk,kk+1; lanes 16-31 K=kk+2,kk+3
            int pa = ((k0 + kk) >> 1) + lh;                 // global k-pair index
            v2f a = *(const v2f*)&Abuf[(pa * 16 + l15) * 2];
            int pb = (kk >> 1) + lh;                        // chunk k-pair index
            v2f b0 = *(const v2f*)&Bbuf[(pb * CP + c0 * 16 + l15) * 2];
            v2f b1 = *(const v2f*)&Bbuf[(pb * CP + c1 * 16 + l15) * 2];
            acc0 = __builtin_amdgcn_wmma_f32_16x16x4_f32(
                false, a, false, b0, (short)0, acc0, false, false);
            acc1 = __builtin_amdgcn_wmma_f32_16x16x4_f32(
                false, a, false, b1, (short)0, acc1, false, false);
        }
    }
    __syncthreads();
    // spill C tiles (row = e + 8*(lane>=16), col = c*16 + lane%16)
#pragma unroll
    for (int e = 0; e < 8; ++e) {
        Grow[(e + 8 * lh) * CP + c0 * 16 + l15] = acc0[e];
        Grow[(e + 8 * lh) * CP + c1 * 16 + l15] = acc1[e];
    }
    __syncthreads();
    // D_ = sqrt(max(sq_s + sq_t - 2G, 0)) * gauss(s,t), in place (t < S only)
    for (int idx = tid; idx < 16 * S; idx += 256) {
        int mr = idx / S, t = idx % S;
        int sg = m0 + mr;
        if (sg < S) {
            float g  = Grow[mr * CP + t];
            float d2 = fmaxf(sq[sg] + sq[t] - 2.0f * g, 0.f);
            float nrm = (d2 > 0.f) ? sqrtf(d2) : 0.f;
            int si = sg / NN, sj = sg % NN;
            int ti = t / NN,  tj = t % NN;
            float dd = (float)((si - ti) * (si - ti) + (sj - tj) * (sj - tj));
            Grow[mr * CP + t] = nrm * expf(-dd / 8.82f);   // 2*(0.15*14)^2 = 8.82
        }
    }
    __syncthreads();
    if (tid < 16) {
        float ssum = 0.f;
        for (int t = 0; t < S; ++t) ssum += Grow[tid * CP + t];
        rowsum[tid] = ssum;
    }
    __syncthreads();
    float* Dws = ws + WS_D + (size_t)b * S * S;
    for (int idx = tid; idx < 16 * S; idx += 256) {
        int mr = idx / S, t = idx % S;
        int sg = m0 + mr;
        if (sg < S) Dws[(size_t)sg * S + t] = Grow[mr * CP + t] / rowsum[mr];
    }
}

// ---------------- kernel 3: one power-iteration step ----------------
__global__ void sp_mv(float* __restrict__ ws, const float* __restrict__ Min,
                      float* __restrict__ Mout, int iter) {
    __shared__ float Ml[S];
    __shared__ float cand[S];
    __shared__ float red[256];
    int b = blockIdx.x, tid = threadIdx.x;
    int wave = tid >> 5, lane = tid & 31;
    if (tid < S) Ml[tid] = Min[b * S + tid];
    __syncthreads();
    const float* Drow = ws + WS_D + (size_t)b * S * S;
    for (int r = wave; r < S; r += 8) {
        float p = 0.f;
        for (int t = lane; t < S; t += 32)
            p += Drow[(size_t)r * S + t] * Ml[t];
        for (int off = 16; off > 0; off >>= 1) p += __shfl_xor(p, off, 32);
        if (lane == 0) cand[r] = p;
    }
    __syncthreads();
    red[tid] = (tid < S) ? fabsf(cand[tid] - Ml[tid]) : 0.f;
    __syncthreads();
    for (int sft = 128; sft > 0; sft >>= 1) {
        if (tid < sft) red[tid] += red[tid + sft];
        __syncthreads();
    }
    int done = ((const int*)(ws + WS_DONE))[iter];
    if (tid < S) Mout[b * S + tid] = done ? Ml[tid] : cand[tid];
    if (tid == 0) ws[WS_ERR + b] = red[0];
}

// ---------------- kernel 4: global convergence flag ----------------
__global__ void sp_flag(float* __restrict__ ws, int iter) {
    float s = 0.f;
    for (int i = 0; i < BB; ++i) s += ws[WS_ERR + i];   // fixed order: deterministic
    float err = s / (float)(BB * S);
    int* done = (int*)(ws + WS_DONE);
    done[iter + 1] = done[iter] | (err < ERR_TH ? 1 : 0);
}

// ---------------- kernel 5: out = M[b,s] * feature ----------------
__global__ void sp_final(const float* __restrict__ f, const float* __restrict__ M,
                         float* __restrict__ out, int total) {
    int idx = blockIdx.x * 256 + threadIdx.x;
    if (idx < total) {
        int b = idx / (KCH * S);
        int s = idx % S;
        out[idx] = f[idx] * M[b * S + s];
    }
}

extern "C" void kernel_launch(void* const* d_in, const int* in_sizes, int n_in,
                              void* d_out, int out_size, void* d_ws, size_t ws_size,
                              hipStream_t stream) {
    const float* f = (const float*)d_in[0];
    float* out = (float*)d_out;
    float* ws  = (float*)d_ws;

    sp_init<<<BB, 256, 0, stream>>>(f, ws);

    size_t shmem = (size_t)(KCH * 16 + KC * CP + 16 * CP + 16) * sizeof(float); // ~114.8 KB
    sp_gram_d<<<dim3(13, BB), 256, shmem, stream>>>(f, ws);

    for (int i = 0; i < MAX_ITER; ++i) {
        const float* Min  = ws + WS_M0 + (i & 1) * (BB * S);
        float*       Mout = ws + WS_M0 + ((i + 1) & 1) * (BB * S);
        sp_mv<<<BB, 256, 0, stream>>>(ws, Min, Mout, i);
        sp_flag<<<1, 1, 0, stream>>>(ws, i);
    }
    // after 10 iterations the current M is back in buffer 0
    sp_final<<<(out_size + 255) / 256, 256, 0, stream>>>(f, ws + WS_M0, out, out_size);
}